// PaperQNN_32590211842587
// MI455X (gfx1250) — compile-verified
//
#include <hip/hip_runtime.h>
#include <math.h>

// MI455X / gfx1250, wave32. Whole-problem restructure:
//   out[b] = MLP( q[b] ),  q[b] = ||P0 Uc s_b||^2 - ||P1 Uc s_b||^2,  y = Uc s_b
// s_b = per-sample encoding state (only sample-dependent circuit part),
// Uc  = fixed classifier unitary built once per launch into d_ws (256KB, L2-resident).
// GEMM Y = S(16x256) x Uc^T on v_wmma_f32_16x16x4_f32 (full fp32 precision).
// Per-tile input x staged into LDS with the Tensor Data Mover (tensor_load_to_lds),
// double-buffered and synchronized with s_wait_tensorcnt (TENSORcnt).
//
// State layout per sample (one wave): flat amp index a[7:0], qubit i <-> flat bit (7-i).
//   lane = a[7:3] (qubits 0..4, cross-lane via shuffles), reg = a[2:0] (qubits 5..7).

#ifndef __has_builtin
#define __has_builtin(x) 0
#endif
#if __has_builtin(__builtin_amdgcn_tensor_load_to_lds)
#define HAVE_TDM 1
#else
#define HAVE_TDM 0
#endif

typedef float v2f  __attribute__((ext_vector_type(2)));
typedef float v8f  __attribute__((ext_vector_type(8)));
typedef unsigned int u32x4 __attribute__((ext_vector_type(4)));
typedef int   i32x4 __attribute__((ext_vector_type(4)));
typedef int   i32x8 __attribute__((ext_vector_type(8)));

#define PI_OVER_8  0.39269908169872414f
#define INV_SQRT2  0.70710678118654752f

__device__ __forceinline__ float lane_shfl(float v, int src)      { return __shfl(v, src, 32); }
__device__ __forceinline__ float lane_shfl_xor(float v, int mask) { return __shfl_xor(v, mask, 32); }

// LDS byte offset of a __shared__ object: flat LDS-aperture address keeps the raw
// LDS offset in addr[31:0] (CDNA5 ISA 10.2 aperture table).
__device__ __forceinline__ unsigned lds_byte_off(const void* p) {
  return (unsigned)(unsigned long long)(uintptr_t)p;
}

// ---- TDM: async-load one x tile (1024 contiguous f32 = 4KB) into LDS ----
__device__ __forceinline__ void tdm_issue_x_tile(const float* __restrict__ gsrc,
                                                 float* ldst, int lane) {
#if HAVE_TDM
  unsigned long long ga = (unsigned long long)(uintptr_t)gsrc;
  u32x4 g0;
  g0[0] = 1u;                                   // count=1 (valid), user mode, no gather
  g0[1] = lds_byte_off(ldst);                   // lds_addr (bytes)
  g0[2] = (unsigned)(ga & 0xffffffffu);         // global_addr[31:0]
  g0[3] = (unsigned)((ga >> 32) & 0x01ffffffu)  // global_addr[56:32]
        | (2u << 30);                           // type=2 ("image")
  i32x8 g1;
  g1[0] = (int)(2u << 16);        // workgroup_mask=0 | data_size=2 (4B)
  g1[1] = (int)(0x0400u << 16);   // tensor_dim0 = 1024 (low16 at [31:16])
  g1[2] = (int)(1u << 16);        // tensor_dim0 hi16=0 | tensor_dim1=1 (low16)
  g1[3] = (int)(0x0400u << 16);   // tensor_dim1 hi16=0 | tile_dim0 = 1024
  g1[4] = 1;                      // tile_dim1=1 | tile_dim2=0
  g1[5] = 1024;                   // tensor_dim0_stride low32
  g1[6] = 0;                      // stride0 hi16 | stride1 low16
  g1[7] = 0;                      // stride1 hi32 (unused, 2D)
  i32x4 z4 = {0, 0, 0, 0};        // group2/group3: zero (2D tensor)
#if defined(__clang_major__) && (__clang_major__ >= 23)
  i32x8 z8 = {0, 0, 0, 0, 0, 0, 0, 0};
  __builtin_amdgcn_tensor_load_to_lds(g0, g1, z4, z4, z8, 0);
#else
  __builtin_amdgcn_tensor_load_to_lds(g0, g1, z4, z4, 0);
#endif
#else
  // Fallback: the DMA wave copies the tile itself.
  for (int i = lane; i < 1024; i += 32) ldst[i] = gsrc[i];
#endif
}

__device__ __forceinline__ void tdm_wait() {
#if HAVE_TDM
  __builtin_amdgcn_s_wait_tensorcnt(0);
#endif
}

// RY(theta) on flat bit P: new_a = c*old_a + s*sgn(a_P)*old_{a^P}
template<int P>
__device__ __forceinline__ void ry_bit(float (&s)[8], float c, float sn, int lane) {
  if constexpr (P < 3) {                       // register-bit qubit
    constexpr int bit = 1 << P;
#pragma unroll
    for (int r = 0; r < 8; ++r) {
      if ((r & bit) == 0) {
        float a0 = s[r], a1 = s[r | bit];
        s[r]       = c * a0 - sn * a1;
        s[r | bit] = sn * a0 + c * a1;
      }
    }
  } else {                                     // lane-bit qubit
    constexpr int mask = 1 << (P - 3);
    float ssg = (lane & mask) ? sn : -sn;
#pragma unroll
    for (int r = 0; r < 8; ++r) {
      float partner = lane_shfl_xor(s[r], mask);
      s[r] = fmaf(c, s[r], ssg * partner);
    }
  }
}

// CNOT: new[a] = old[a ^ (bit_PC(a) << PT)]
template<int PC, int PT>
__device__ __forceinline__ void cnot_bits(float (&s)[8], int lane) {
  if constexpr (PC >= 3 && PT >= 3) {          // lane permutation
    constexpr int cm = 1 << (PC - 3);
    constexpr int tm = 1 << (PT - 3);
    int src = lane ^ ((lane & cm) ? tm : 0);
#pragma unroll
    for (int r = 0; r < 8; ++r) s[r] = lane_shfl(s[r], src);
  } else if constexpr (PC >= 3 && PT < 3) {    // lane-conditional register swap
    constexpr int cm = 1 << (PC - 3);
    constexpr int tb = 1 << PT;
    bool ctrl = (lane & cm) != 0;
#pragma unroll
    for (int r = 0; r < 8; ++r) {
      if ((r & tb) == 0) {
        float a = s[r], b = s[r | tb];
        s[r]      = ctrl ? b : a;
        s[r | tb] = ctrl ? a : b;
      }
    }
  } else if constexpr (PC < 3 && PT < 3) {     // pure register rename (free)
    constexpr int cb = 1 << PC;
    constexpr int tb = 1 << PT;
#pragma unroll
    for (int r = 0; r < 8; ++r) {
      if ((r & cb) != 0 && (r & tb) == 0) {
        float t = s[r]; s[r] = s[r | tb]; s[r | tb] = t;
      }
    }
  } else {                                     // register control, lane target
    constexpr int cb = 1 << PC;
    constexpr int tm = 1 << (PT - 3);
#pragma unroll
    for (int r = 0; r < 8; ++r)
      if ((r & cb) != 0) s[r] = lane_shfl_xor(s[r], tm);
  }
}

// Encoding: H(q0..q5) + layer-0 RYs folded into a product-state init (pure VALU,
// saves 5 cross-lane RYs = 40 ds_bpermute per sample), then CNOT chain + layers 1..3.
__device__ __forceinline__ void encode_sample(const float* __restrict__ xrow,
                                              float (&s)[8], int lane) {
  // lane (l=lane>>3, i=lane&7) owns angle (layer l, qubit i): x index l*16+i
  float xv = xrow[((lane >> 3) << 4) | (lane & 7)];
  float th = (xv + 1.0f) * PI_OVER_8;          // theta/2 = (x+1)*pi/8
  float cl = __cosf(th);
  float sl = __sinf(th);

  // ---- product-state init: amp(a) = prod_i f_i(bit_i(a)) ----
  // qubits 0..5 (H'd): f(0)=(c-s)/sqrt2, f(1)=(c+s)/sqrt2; qubits 6,7: f(0)=c, f(1)=s
  float L = 1.0f;
#pragma unroll
  for (int i = 0; i < 5; ++i) {                // qubits 0..4 live on lane bits 4..0
    float ci = lane_shfl(cl, i), si = lane_shfl(sl, i);
    float f = ((lane >> (4 - i)) & 1) ? (ci + si) : (ci - si);
    L *= f * INV_SQRT2;
  }
  float c5 = lane_shfl(cl, 5), s5 = lane_shfl(sl, 5);
  float c6 = lane_shfl(cl, 6), s6 = lane_shfl(sl, 6);
  float c7 = lane_shfl(cl, 7), s7 = lane_shfl(sl, 7);
  float f5a = (c5 - s5) * INV_SQRT2, f5b = (c5 + s5) * INV_SQRT2;
#pragma unroll
  for (int r = 0; r < 8; ++r) {                // reg bits 2,1,0 = qubits 5,6,7
    float f5 = (r & 4) ? f5b : f5a;
    float f6 = (r & 2) ? s6 : c6;
    float f7 = (r & 1) ? s7 : c7;
    s[r] = L * f5 * f6 * f7;
  }
  // layer-0 CNOT chain
  cnot_bits<7,6>(s,lane); cnot_bits<6,5>(s,lane); cnot_bits<5,4>(s,lane);
  cnot_bits<4,3>(s,lane); cnot_bits<3,2>(s,lane); cnot_bits<2,1>(s,lane);
  cnot_bits<1,0>(s,lane);

#define ENC_RY(LY, I) { float c_ = lane_shfl(cl, (LY)*8+(I)); \
                        float n_ = lane_shfl(sl, (LY)*8+(I)); \
                        ry_bit<7-(I)>(s, c_, n_, lane); }
#define ENC_LAYER(LY)                                                     \
  ENC_RY(LY,0) ENC_RY(LY,1) ENC_RY(LY,2) ENC_RY(LY,3)                     \
  ENC_RY(LY,4) ENC_RY(LY,5) ENC_RY(LY,6) ENC_RY(LY,7)                     \
  cnot_bits<7,6>(s,lane); cnot_bits<6,5>(s,lane); cnot_bits<5,4>(s,lane); \
  cnot_bits<4,3>(s,lane); cnot_bits<3,2>(s,lane); cnot_bits<2,1>(s,lane); \
  cnot_bits<1,0>(s,lane);
  ENC_LAYER(1) ENC_LAYER(2) ENC_LAYER(3)
#undef ENC_RY
#undef ENC_LAYER
}

// ---------------- Kernel 1: build Uc (256x256), stored row-major U[a][j] ----------------
__global__ __launch_bounds__(32)
void qnn_build_unitary(const float* __restrict__ cparams, float* __restrict__ U) {
  const int lane = threadIdx.x;      // 32 threads = 1 wave
  const int j = blockIdx.x;          // basis column 0..255
  float th = cparams[lane] * 0.5f;   // lane owns cparams[lane]
  float cc = __cosf(th), ss = __sinf(th);
  float s[8];
#pragma unroll
  for (int r = 0; r < 8; ++r)
    s[r] = ((lane == (j >> 3)) && (r == (j & 7))) ? 1.0f : 0.0f;
#define CLS_STEP(P, I) { float c_ = lane_shfl(cc, (P)); float n_ = lane_shfl(ss, (P)); \
    ry_bit<7-(I)>(s, c_, n_, lane); ry_bit<6-(I)>(s, c_, n_, lane);                    \
    cnot_bits<7-(I), 6-(I)>(s, lane); }
#define CLS_DEPTH(D)                                                                   \
  CLS_STEP((D)*8+0,0) CLS_STEP((D)*8+1,1) CLS_STEP((D)*8+2,2) CLS_STEP((D)*8+3,3)     \
  CLS_STEP((D)*8+4,4) CLS_STEP((D)*8+5,5) CLS_STEP((D)*8+6,6)                          \
  { float c_ = lane_shfl(cc, (D)*8+7); float n_ = lane_shfl(ss, (D)*8+7);              \
    ry_bit<0>(s, c_, n_, lane); ry_bit<7>(s, c_, n_, lane);                            \
    cnot_bits<0,7>(s, lane); }
  CLS_DEPTH(0) CLS_DEPTH(1) CLS_DEPTH(2) CLS_DEPTH(3)
#undef CLS_STEP
#undef CLS_DEPTH
#pragma unroll
  for (int r = 0; r < 8; ++r)
    U[(lane * 8 + r) * 256 + j] = s[r];   // U[a][j], a = lane*8+r
}

// -------- Kernel 2: persistent WGs; TDM-staged x; encode; WMMA vs Uc; reduce; MLP --------
__global__ __launch_bounds__(128)
void qnn_main(const float* __restrict__ x, const float* __restrict__ U,
              const float* __restrict__ w1, const float* __restrict__ b1,
              const float* __restrict__ w2, const float* __restrict__ b2,
              float* __restrict__ out, int ntiles) {
  __shared__ float xbuf[2][1024];     // double-buffered x tiles (16 samples x 64)
  __shared__ float Stile[16 * 256];   // 16 encoded states
  __shared__ float qbuf[16];

  const int tid  = threadIdx.x;
  const int lane = tid & 31;
  const int wave = tid >> 5;          // 0..3 (wave 0 is the DMA wave)
  const int stride = (int)gridDim.x;

  int tile = (int)blockIdx.x;
  if (wave == 0 && tile < ntiles)
    tdm_issue_x_tile(x + (size_t)tile * 1024, &xbuf[0][0], lane);

  int cur = 0;
  for (; tile < ntiles; tile += stride, cur ^= 1) {
    if (wave == 0) tdm_wait();        // xbuf[cur] landed (TENSORcnt in-order per wave)
    __syncthreads();                  // publish xbuf[cur]; prev-iter LDS reads all done

    if (tid < 16) qbuf[tid] = 0.0f;

    const int ntile = tile + stride;  // prefetch next tile into the other buffer
    if (wave == 0 && ntile < ntiles)
      tdm_issue_x_tile(x + (size_t)ntile * 1024, &xbuf[cur ^ 1][0], lane);

    // ---- encoding: 4 samples per wave, states into Stile ----
#pragma unroll 1
    for (int k = 0; k < 4; ++k) {
      const int m = wave * 4 + k;
      float s[8];
      encode_sample(&xbuf[cur][m * 64], s, lane);
      float4* dst = (float4*)&Stile[m * 256 + lane * 8];
      dst[0] = make_float4(s[0], s[1], s[2], s[3]);
      dst[1] = make_float4(s[4], s[5], s[6], s[7]);
    }
    __syncthreads();

    // ---- Y = S x Uc^T via v_wmma_f32_16x16x4_f32; wave w owns N-tiles 4w..4w+3 ----
    // A frag (16x4): lanes 0-15 -> M=lane, K={kk,kk+1}; lanes 16-31 -> K={kk+2,kk+3}
    // B frag (4x16): lanes 0-15 -> N=lane, rows {kk,kk+1}; lanes 16-31 -> {kk+2,kk+3}
    const int Mrow = lane & 15;
    const int koff = (lane >> 4) << 1;
    const int t0   = wave * 4;

    v8f acc0 = {0,0,0,0,0,0,0,0};
    v8f acc1 = acc0, acc2 = acc0, acc3 = acc0;

#pragma unroll 4
    for (int kk = 0; kk < 256; kk += 4) {
      v2f A;
      A.x = Stile[Mrow * 256 + kk + koff];
      A.y = Stile[Mrow * 256 + kk + koff + 1];
      const float* u0 = U + ((t0 + 0) * 16 + Mrow) * 256 + kk + koff;
      const float* u1 = u0 + 16 * 256;
      const float* u2 = u1 + 16 * 256;
      const float* u3 = u2 + 16 * 256;
      v2f B0; B0.x = u0[0]; B0.y = u0[1];
      v2f B1; B1.x = u1[0]; B1.y = u1[1];
      v2f B2; B2.x = u2[0]; B2.y = u2[1];
      v2f B3; B3.x = u3[0]; B3.y = u3[1];
      acc0 = __builtin_amdgcn_wmma_f32_16x16x4_f32(false, A, false, B0, (short)0, acc0, false, false);
      acc1 = __builtin_amdgcn_wmma_f32_16x16x4_f32(false, A, false, B1, (short)0, acc1, false, false);
      acc2 = __builtin_amdgcn_wmma_f32_16x16x4_f32(false, A, false, B2, (short)0, acc2, false, false);
      acc3 = __builtin_amdgcn_wmma_f32_16x16x4_f32(false, A, false, B3, (short)0, acc3, false, false);
    }

    // ---- fused measurement: q = sum_{a<128} y^2 - sum_{a>=128} y^2 ----
    // a = 16*(t0+ti)+N; waves 0,1 cover a<128 (+), waves 2,3 a>=128 (-)
    const float sgn = (wave < 2) ? 1.0f : -1.0f;
    float psum[8];
#pragma unroll
    for (int r = 0; r < 8; ++r)
      psum[r] = sgn * (acc0[r]*acc0[r] + acc1[r]*acc1[r] + acc2[r]*acc2[r] + acc3[r]*acc3[r]);

    // reduce over N within each 16-lane half (D frag: M = r + 8*(lane>=16), N = lane&15)
#pragma unroll
    for (int msk = 1; msk < 16; msk <<= 1)
#pragma unroll
      for (int r = 0; r < 8; ++r)
        psum[r] += lane_shfl_xor(psum[r], msk);

    if ((lane & 15) == 0) {
      const int mb = (lane >> 4) << 3;   // lane 0 -> samples 0..7, lane 16 -> 8..15
#pragma unroll
      for (int r = 0; r < 8; ++r) atomicAdd(&qbuf[mb + r], psum[r]);  // ds_add_f32
    }
    __syncthreads();

    // ---- MLP: out = b2 + sum_j w2[j]*relu(q*w1[j]+b1[j]) ----
    if (tid < 16) {
      float q = qbuf[tid];
      float o = b2[0];
#pragma unroll
      for (int jj = 0; jj < 16; ++jj)
        o += w2[jj] * fmaxf(fmaf(q, w1[jj], b1[jj]), 0.0f);
      out[(size_t)tile * 16 + tid] = o;
    }
    // next iteration's leading __syncthreads orders qbuf/Stile/xbuf reuse
  }
}

extern "C" void kernel_launch(void* const* d_in, const int* in_sizes, int n_in,
                              void* d_out, int out_size, void* d_ws, size_t ws_size,
                              hipStream_t stream) {
  const float* x  = (const float*)d_in[0];   // (B,1,8,8) fp32
  const float* cp = (const float*)d_in[1];   // (32,)
  const float* w1 = (const float*)d_in[2];   // (16,1)
  const float* b1 = (const float*)d_in[3];   // (16,)
  const float* w2 = (const float*)d_in[4];   // (1,16)
  const float* b2 = (const float*)d_in[5];   // (1,)
  float* U = (float*)d_ws;                   // 256*256*4 = 256KB scratch for Uc
  const int B = in_sizes[0] / 64;            // 131072
  const int ntiles = B / 16;                 // 8192

  qnn_build_unitary<<<256, 32, 0, stream>>>(cp, U);
  int grid = ntiles < 2048 ? ntiles : 2048;  // persistent WGs, ~4 tiles each
  qnn_main<<<grid, 128, 0, stream>>>(x, U, w1, b1, w2, b2, (float*)d_out, ntiles);
}